// LEFTRelationEMB_18270790877545
// MI455X (gfx1250) — compile-verified
//
#include <hip/hip_runtime.h>
#include <hip/hip_bf16.h>

// ---------------------------------------------------------------------------
// LEFTRelationEMB on MI455X (gfx1250), all-f32 pipeline on V_WMMA_F32_16X16X4_F32.
//
// Sizes: N=12 (144 pairs), C=256, H=W=32 (HW=1024), C3=384, CF=640, OC=128,
//        DIN=131072, OUT=128.
// ---------------------------------------------------------------------------

typedef float v2f __attribute__((ext_vector_type(2)));
typedef float v8f __attribute__((ext_vector_type(8)));

#define WMMA_F32(a, b, c)                                                      \
  __builtin_amdgcn_wmma_f32_16x16x4_f32(false, (a), false, (b), (short)0, (c), \
                                        false, false)

#define NOBJ   12
#define NPAIR  144
#define CIN    256
#define HW     1024
#define WIMG   32
#define C3     384
#define CF     640
#define OC     128
#define DIN    131072   // 128*1024
#define OUTD   128
#define SCALEF (1.0f / 16.0f)

// Column-major sB row stride (dwords): 640 + 4 pad -> conflict-free b64 reads.
#define SBSTR  644

// ---------------------------------------------------------------------------
// Kernel 0: rel_feat[384][1024] = w_extract[384][256] @ scene[256][1024] + b
// 24 M-tiles x 64 N-tiles = 1536 waves; one 16x16 tile per wave, K=256.
// ---------------------------------------------------------------------------
__global__ __launch_bounds__(256) void k_extract(const float* __restrict__ scene,
                                                 const float* __restrict__ wext,
                                                 const float* __restrict__ bext,
                                                 float* __restrict__ rel) {
  const int lane = threadIdx.x & 31;
  const int l16  = lane & 15;
  const int kk   = lane >> 4;                   // 0 or 1
  const int wave = blockIdx.x * 8 + (threadIdx.x >> 5);  // 0..1535
  const int tm = wave >> 6;                     // 0..23
  const int tn = wave & 63;                     // 0..63
  const int row = tm * 16 + l16;                // A row (0..383)
  const int col = tn * 16 + l16;                // B/D col (0..1023)

  v8f acc = {};
  const float* __restrict__ ap = wext + row * CIN + 2 * kk;
  for (int kb = 0; kb < CIN; kb += 4) {
    v2f a = *(const v2f*)(ap + kb);
    v2f b;
    const int kr = kb + 2 * kk;
    b.x = scene[(kr + 0) * HW + col];
    b.y = scene[(kr + 1) * HW + col];
    acc = WMMA_F32(a, b, acc);
  }
#pragma unroll
  for (int v = 0; v < 8; ++v) {
    const int m = tm * 16 + v + 8 * kk;
    rel[m * HW + col] = acc[v] + bext[m];
  }
}

// ---------------------------------------------------------------------------
// Kernel 1: one workgroup (256 thr = 8 waves) per pair.
//  - geometry (union box, bilinear taps, imap coverage) in LDS
//  - builds the 640-row concat input, 16 spatial columns at a time, in LDS
//    (column-major + pad so B fragments are single ds_load_b64, bank-clean)
//  - fused conv GEMM [128x640]@[640x16] per chunk: wave w owns M-tile w
//  - relu + bias, writes fused activations to workspace
// ---------------------------------------------------------------------------
__global__ __launch_bounds__(256) void k_pair(const float* __restrict__ box,
                                              const float* __restrict__ objf,
                                              const float* __restrict__ rel,
                                              const float* __restrict__ wfuse,
                                              const float* __restrict__ bfuse,
                                              float* __restrict__ fused) {
  __shared__ float sB[16 * SBSTR];   // ~40.3 KB: 16 cols x (640 rows + pad)
  __shared__ int   sx0[32], sx1[32], sy0[32], sy1[32];
  __shared__ float swx[32], swy[32];
  __shared__ float scxs[32], scys[32], scxo[32], scyo[32];

  const int p   = blockIdx.x;       // pair id
  const int sid = p / NOBJ;
  const int oid = p % NOBJ;
  const int tid = threadIdx.x;
  const int lane = tid & 31;
  const int l16  = lane & 15;
  const int kk   = lane >> 4;
  const int wv   = tid >> 5;        // wave in block = M-tile

  if (tid < 32) {
    const int k = tid;
    const float s_x1 = box[sid * 4 + 0], s_y1 = box[sid * 4 + 1];
    const float s_x2 = box[sid * 4 + 2], s_y2 = box[sid * 4 + 3];
    const float o_x1 = box[oid * 4 + 0], o_y1 = box[oid * 4 + 1];
    const float o_x2 = box[oid * 4 + 2], o_y2 = box[oid * 4 + 3];
    const float ux1 = fminf(s_x1, o_x1), uy1 = fminf(s_y1, o_y1);
    const float ux2 = fmaxf(s_x2, o_x2), uy2 = fmaxf(s_y2, o_y2);

    // roi_align sample coords (aligned=true, one sample per bin center)
    const float bx1 = ux1 * SCALEF - 0.5f, bx2 = ux2 * SCALEF - 0.5f;
    const float by1 = uy1 * SCALEF - 0.5f, by2 = uy2 * SCALEF - 0.5f;
    const float bw = (bx2 - bx1) * (1.0f / 32.0f);
    const float bh = (by2 - by1) * (1.0f / 32.0f);
    float xc = fminf(fmaxf(bx1 + ((float)k + 0.5f) * bw, 0.0f), 31.0f);
    float yc = fminf(fmaxf(by1 + ((float)k + 0.5f) * bh, 0.0f), 31.0f);
    const int xi0 = (int)floorf(xc);
    const int yi0 = (int)floorf(yc);
    sx0[k] = xi0; sx1[k] = min(xi0 + 1, 31); swx[k] = xc - (float)xi0;
    sy0[k] = yi0; sy1[k] = min(yi0 + 1, 31); swy[k] = yc - (float)yi0;

    // imap coverage factors (image coords over union grid)
    const float cellx = (ux2 - ux1) * (1.0f / 32.0f);
    const float clox  = ux1 + (float)k * cellx;
    const float icx   = 1.0f / fmaxf(cellx, 1e-6f);
    scxs[k] = fmaxf(fminf(clox + cellx, s_x2) - fmaxf(clox, s_x1), 0.0f) * icx;
    scxo[k] = fmaxf(fminf(clox + cellx, o_x2) - fmaxf(clox, o_x1), 0.0f) * icx;
    const float celly = (uy2 - uy1) * (1.0f / 32.0f);
    const float cloy  = uy1 + (float)k * celly;
    const float icy   = 1.0f / fmaxf(celly, 1e-6f);
    scys[k] = fmaxf(fminf(cloy + celly, s_y2) - fmaxf(cloy, s_y1), 0.0f) * icy;
    scyo[k] = fmaxf(fminf(cloy + celly, o_y2) - fmaxf(cloy, o_y1), 0.0f) * icy;
  }
  __syncthreads();

  for (int ch = 0; ch < 64; ++ch) {
    const int colBase = ch * 16;

    // Build 640x16 input chunk (column-major): rows 0-127 objf[sub],
    // 128-255 objf[obj], 256-639 roi-aligned rel_feat (x | y*sub | z*obj).
    for (int i = tid; i < CF * 16; i += 256) {
      const int r = i >> 4;
      const int cidx = i & 15;
      const int s = colBase + cidx;
      float val;
      if (r < 128) {
        val = objf[sid * DIN + r * HW + s];
      } else if (r < 256) {
        val = objf[oid * DIN + (r - 128) * HW + s];
      } else {
        const int c  = r - 256;            // pooled channel 0..383
        const int py = s >> 5;
        const int px = s & 31;
        const float* f = rel + c * HW;
        const int y0 = sy0[py], y1 = sy1[py];
        const int x0 = sx0[px], x1 = sx1[px];
        const float wxv = swx[px], wyv = swy[py];
        const float v00 = f[y0 * WIMG + x0];
        const float v01 = f[y0 * WIMG + x1];
        const float v10 = f[y1 * WIMG + x0];
        const float v11 = f[y1 * WIMG + x1];
        val = (1.0f - wyv) * ((1.0f - wxv) * v00 + wxv * v01) +
              wyv * ((1.0f - wxv) * v10 + wxv * v11);
        if (c >= 256)      val *= scyo[py] * scxo[px];
        else if (c >= 128) val *= scys[py] * scxs[px];
      }
      sB[cidx * SBSTR + r] = val;
    }
    __syncthreads();

    // GEMM: wave wv computes output rows [16*wv, 16*wv+16) x 16 cols, K=640.
    const int mrow = wv * 16 + l16;
    const float* __restrict__ ap = wfuse + mrow * CF + 2 * kk;
    const float* __restrict__ bp = &sB[l16 * SBSTR + 2 * kk];
    v8f acc = {};
    for (int kb = 0; kb < CF; kb += 4) {
      v2f a = *(const v2f*)(ap + kb);      // global_load_b64
      v2f b = *(const v2f*)(bp + kb);      // ds_load_b64, bank-conflict-free
      acc = WMMA_F32(a, b, acc);
    }
#pragma unroll
    for (int v = 0; v < 8; ++v) {
      const int m = wv * 16 + v + 8 * kk;
      const float r = acc[v] + bfuse[m];
      fused[p * DIN + m * HW + colBase + l16] = fmaxf(r, 0.0f);
    }
    __syncthreads();
  }
}

// ---------------------------------------------------------------------------
// Kernel 2: initialize emb accumulator with the fc bias.
// ---------------------------------------------------------------------------
__global__ __launch_bounds__(256) void k_embinit(const float* __restrict__ bfc,
                                                 float* __restrict__ emb) {
  const int i = blockIdx.x * 256 + threadIdx.x;
  if (i < NPAIR * OUTD) emb[i] = bfc[i & (OUTD - 1)];
}

// ---------------------------------------------------------------------------
// Kernel 3: emb[144][128] += relu_flat[144][131072] @ w_fc.T, split-K.
// 9 M-tiles x 8 N-tiles x 64 K-chunks (2048 each) = 4608 waves.
// Streams are sequential; prefetch 256 floats (two 128-float sub-blocks) ahead.
// ---------------------------------------------------------------------------
__global__ __launch_bounds__(256) void k_fc(const float* __restrict__ fused,
                                            const float* __restrict__ wfc,
                                            float* __restrict__ emb) {
  const int lane = threadIdx.x & 31;
  const int l16  = lane & 15;
  const int kk   = lane >> 4;
  const int wid  = blockIdx.x * 8 + (threadIdx.x >> 5);   // 0..4607
  const int tm  = wid / 512;          // 0..8
  const int rem = wid % 512;
  const int tn  = rem / 64;           // 0..7
  const int kc  = rem % 64;           // K-chunk id
  const int arow = tm * 16 + l16;     // pair row 0..143
  const int bcol = tn * 16 + l16;     // output channel 0..127
  const long kbase = (long)kc * 2048;

  v8f acc = {};
  const float* __restrict__ ap = fused + (long)arow * DIN + kbase + 2 * kk;
  const float* __restrict__ bp = wfc + (long)bcol * DIN + kbase + 2 * kk;
  for (int sb = 0; sb < 2048; sb += 128) {
    __builtin_prefetch(ap + sb + 256, 0, 1);   // global_prefetch_b8
    __builtin_prefetch(bp + sb + 256, 0, 1);
#pragma unroll
    for (int kb = 0; kb < 128; kb += 4) {
      v2f a = *(const v2f*)(ap + sb + kb);     // global_load_b64
      v2f b = *(const v2f*)(bp + sb + kb);     // global_load_b64
      acc = WMMA_F32(a, b, acc);
    }
  }
#pragma unroll
  for (int v = 0; v < 8; ++v) {
    const int m = tm * 16 + v + 8 * kk;
    atomicAdd(&emb[m * OUTD + bcol], acc[v]);
  }
}

// ---------------------------------------------------------------------------
// Kernel 4: L2-normalize each 128-wide row into d_out.
// ---------------------------------------------------------------------------
__global__ __launch_bounds__(128) void k_norm(const float* __restrict__ emb,
                                              float* __restrict__ out) {
  __shared__ float red[128];
  const int row = blockIdx.x;
  const int t = threadIdx.x;
  const float v = emb[row * OUTD + t];
  red[t] = v * v;
  __syncthreads();
  for (int s = 64; s > 0; s >>= 1) {
    if (t < s) red[t] += red[t + s];
    __syncthreads();
  }
  const float inv = rsqrtf(red[0]);
  out[row * OUTD + t] = v * inv;
}

// ---------------------------------------------------------------------------
// Launcher.  Workspace layout (floats):
//   rel_feat : [0, 393216)
//   fused    : [393216, 393216 + 18874368)
//   emb_acc  : next 18432
// Total ~77.2 MB.
// ---------------------------------------------------------------------------
extern "C" void kernel_launch(void* const* d_in, const int* in_sizes, int n_in,
                              void* d_out, int out_size, void* d_ws, size_t ws_size,
                              hipStream_t stream) {
  (void)in_sizes; (void)n_in; (void)out_size; (void)ws_size;
  const float* scene = (const float*)d_in[0];
  const float* box   = (const float*)d_in[1];
  const float* objf  = (const float*)d_in[2];
  const float* wext  = (const float*)d_in[3];
  const float* bext  = (const float*)d_in[4];
  const float* wfuse = (const float*)d_in[5];
  const float* bfuse = (const float*)d_in[6];
  const float* wfc   = (const float*)d_in[7];
  const float* bfc   = (const float*)d_in[8];

  float* ws    = (float*)d_ws;
  float* rel   = ws;                         // 384*1024
  float* fused = ws + 393216;                // 144*128*1024
  float* emb   = fused + 18874368;           // 144*128

  float* out = (float*)d_out;

  k_extract<<<192, 256, 0, stream>>>(scene, wext, bext, rel);
  k_pair   <<<NPAIR, 256, 0, stream>>>(box, objf, rel, wfuse, bfuse, fused);
  k_embinit<<<72, 256, 0, stream>>>(bfc, emb);
  k_fc     <<<576, 256, 0, stream>>>(fused, wfc, emb);
  k_norm   <<<NPAIR, 128, 0, stream>>>(emb, out);
}